// LambdaRankLoss_63874753626649
// MI455X (gfx1250) — compile-verified
//
#include <hip/hip_runtime.h>
#include <hip/hip_bf16.h>
#include <math.h>

// ---------------------------------------------------------------------------
// LambdaRank loss, B=32, N=1024, K=3, sigma=1  (MI455X / gfx1250, wave32).
//
// Roofline: inputs are 256 KB total (L2-resident trivially at 192 MB); the
// cost is the softplus chain (v_exp_f32 + v_log_f32) over the pair space.
// Optimizations this round:
//   * symmetry: loss(i,j)+loss(j,i) = |x| + 2*ln(1+e^-|x|) - s*x  (s=+-1),
//     so only the upper triangle is evaluated -> 2x fewer trans ops.
//   * raw v_exp_f32/v_log_f32 via __builtin_amdgcn_exp2f/logf (1+e is never
//     denormal, so the libm denorm fixups are dead weight).
//   * 16x16 diff tiles via V_WMMA_F32_16X16X4_F32 rank-1 outer product:
//     D = A(p_i) x B(ones) + C(-q_j), result lands in C/D layout directly.
//   * deterministic two-stage reduction (no float atomics).
// ---------------------------------------------------------------------------

typedef __attribute__((ext_vector_type(2))) float v2f;
typedef __attribute__((ext_vector_type(8))) float v8f;

#define BATCH 32
#define NITEMS 1024

// --------------------------- kernel 1: top-3 weights ------------------------
__global__ __launch_bounds__(32) void topk_mult_kernel(const float* __restrict__ pred,
                                                       float* __restrict__ mult) {
  const int b = blockIdx.x;
  const int lane = threadIdx.x;
  const float* p = pred + b * NITEMS;

  int c0 = -1, c1 = -1, c2 = -1;
  for (int pass = 0; pass < 3; ++pass) {
    float bv = -INFINITY;
    int bi = 0x7fffffff;
    for (int i = lane; i < NITEMS; i += 32) {
      if (i == c0 || i == c1 || i == c2) continue;
      float v = p[i];
      if (v > bv || (v == bv && i < bi)) { bv = v; bi = i; }
    }
    // wave32 reduce: max value, ties -> smaller index (lax.top_k order)
    for (int m = 16; m > 0; m >>= 1) {
      float ov = __shfl_xor(bv, m, 32);
      int   oi = __shfl_xor(bi, m, 32);
      if (ov > bv || (ov == bv && oi < bi)) { bv = ov; bi = oi; }
    }
    if (pass == 0) c0 = bi; else if (pass == 1) c1 = bi; else c2 = bi;
  }
  for (int i = lane; i < NITEMS; i += 32) {
    mult[b * NITEMS + i] = (i == c0 || i == c1 || i == c2) ? 2.0f : 1.0f;
  }
}

// --------------------------- kernel 2: pairwise loss ------------------------
// grid = BATCH*8 blocks; block = 256 threads = 8 waves.
// batch = blockIdx/8, chunk = blockIdx%8; wave w owns i-tile (w*8 + chunk)
// (interleaved so triangular trip counts balance across blocks) and loops
// over j-tiles jt = tile..63 (upper triangle only).
__global__ __launch_bounds__(256) void pair_loss_kernel(const float* __restrict__ pred,
                                                        const float* __restrict__ rel,
                                                        const float* __restrict__ mult,
                                                        float* __restrict__ partials) {
  __shared__ __align__(16) float p_s[NITEMS];
  __shared__ __align__(16) float r_s[NITEMS];
  __shared__ __align__(16) float w_s[NITEMS];
  __shared__ float racc[8];
  __shared__ float rcnt[8];

  const int b     = blockIdx.x >> 3;
  const int chunk = blockIdx.x & 7;
  const int tid   = threadIdx.x;

  // stage the whole batch row into LDS (256 threads x float4 = 1024 floats)
  const float4* pg = (const float4*)(pred + b * NITEMS);
  const float4* rg = (const float4*)(rel  + b * NITEMS);
  const float4* wg = (const float4*)(mult + b * NITEMS);
  __builtin_prefetch(rg + tid, 0, 0);   // gfx1250 global_prefetch_b8
  __builtin_prefetch(wg + tid, 0, 0);
  ((float4*)p_s)[tid] = pg[tid];
  ((float4*)r_s)[tid] = rg[tid];
  ((float4*)w_s)[tid] = wg[tid];
  __syncthreads();

  const int lane    = tid & 31;
  const int wave    = tid >> 5;
  const int tile    = wave * 8 + chunk;     // 0..63, interleaved for balance
  const int rowBase = tile * 16;

  // Row-side (i) values for this lane's 8 C/D elements:
  //   C/D layout: VGPR v, lane L -> row = v + 8*(L>=16), col = L&15
  const int rsel = rowBase + ((lane >> 4) << 3);
  float rrow[8], wrow[8];
#pragma unroll
  for (int v = 0; v < 8; ++v) { rrow[v] = r_s[rsel + v]; wrow[v] = w_s[rsel + v]; }

  // A (16x4 f32): A[i][0] = p_i, rest 0 (lanes 0-15 = M rows, K0 in VGPR0).
  v2f a;
  a.x = (lane < 16) ? p_s[rowBase + lane] : 0.0f;
  a.y = 0.0f;
  // B (4x16 f32): B[0][j] = 1, rest 0 (row K=0 in VGPR0 lanes 0-15).
  v2f bones;
  bones.x = (lane < 16) ? 1.0f : 0.0f;
  bones.y = 0.0f;

  const float LOG2E  = 1.4426950408889634f;   // 1/ln2
  const float TWOLN2 = 1.3862943611198906f;   // 2*ln2

  float acc = 0.0f, cnt = 0.0f;

  // One 16x16 tile of unordered pairs; diag==true masks to row < col.
  auto do_tile = [&](int jt, bool diag) {
    const int col = jt * 16 + (lane & 15);
    const float qc = p_s[col];
    const float rc = r_s[col];
    const float wc = w_s[col];

    v8f c;
#pragma unroll
    for (int k = 0; k < 8; ++k) c[k] = -qc;

    // D[i][j] = p_i - q_j on the matrix pipe (co-executes with trans chain)
    v8f d = __builtin_amdgcn_wmma_f32_16x16x4_f32(
        false, a, false, bones, (short)0, c, false, false);

#pragma unroll
    for (int v = 0; v < 8; ++v) {
      const float x  = d[v];
      const float rr = rrow[v];
      float mask = (rr != rc) ? 1.0f : 0.0f;
      if (diag) mask = ((rsel + v) < col) ? mask : 0.0f;
      const float s  = (rr > rc) ? 1.0f : -1.0f;
      const float ax = fabsf(x);
      // pairsum = |x| - s*x + 2*ln(1 + exp(-|x|)) ; raw exp2/log2 hardware ops
      const float e  = __builtin_amdgcn_exp2f(-ax * LOG2E);
      const float l2 = __builtin_amdgcn_logf(1.0f + e);   // log2, arg in (1,2]
      acc += mask * (wrow[v] * wc) * ((ax - s * x) + TWOLN2 * l2);
      cnt += 2.0f * mask;   // unordered pair -> 2 ordered valid pairs
    }
  };

  do_tile(tile, true);                       // diagonal tile: strict upper part
  for (int jt = tile + 1; jt < 64; ++jt) {   // off-diagonal tiles
    do_tile(jt, false);
  }

  // wave32 reduce
  for (int m = 16; m > 0; m >>= 1) {
    acc += __shfl_xor(acc, m, 32);
    cnt += __shfl_xor(cnt, m, 32);
  }
  if (lane == 0) { racc[wave] = acc; rcnt[wave] = cnt; }
  __syncthreads();

  if (tid == 0) {
    float sa = 0.0f, sc = 0.0f;
    for (int wv = 0; wv < 8; ++wv) { sa += racc[wv]; sc += rcnt[wv]; }
    partials[blockIdx.x * 2 + 0] = sa;
    partials[blockIdx.x * 2 + 1] = sc;
  }
}

// --------------------------- kernel 3: finalize -----------------------------
__global__ __launch_bounds__(32) void finalize_kernel(const float* __restrict__ partials,
                                                      float* __restrict__ out) {
  const int b = threadIdx.x;  // 0..31, one batch per lane
  float sa = 0.0f, sc = 0.0f;
  for (int k = 0; k < 8; ++k) {
    sa += partials[(b * 8 + k) * 2 + 0];
    sc += partials[(b * 8 + k) * 2 + 1];
  }
  float loss = sa / (sc + 1e-8f);
  for (int m = 16; m > 0; m >>= 1) loss += __shfl_xor(loss, m, 32);
  if (b == 0) out[0] = loss * (1.0f / (float)BATCH);
}

// --------------------------- launcher ---------------------------------------
extern "C" void kernel_launch(void* const* d_in, const int* in_sizes, int n_in,
                              void* d_out, int out_size, void* d_ws, size_t ws_size,
                              hipStream_t stream) {
  const float* pred = (const float*)d_in[0];
  const float* rel  = (const float*)d_in[1];

  float* ws       = (float*)d_ws;
  float* mult     = ws;                     // BATCH*NITEMS floats (128 KB)
  float* partials = ws + BATCH * NITEMS;    // BATCH*8*2 floats

  topk_mult_kernel<<<BATCH, 32, 0, stream>>>(pred, mult);
  pair_loss_kernel<<<BATCH * 8, 256, 0, stream>>>(pred, rel, mult, partials);
  finalize_kernel<<<1, 32, 0, stream>>>(partials, (float*)d_out);
}